// ScreenHashLoss_53919019434147
// MI455X (gfx1250) — compile-verified
//
#include <hip/hip_runtime.h>

#define BDIM 8192
#define KDIM 128
#define BT 64
#define STRIDE 132                 // pad: conflict-free ds_load_b64, keeps 16B alignment for b128 stores
#define NTILE ((BDIM / BT) * (BDIM / BT))

typedef float v2f __attribute__((ext_vector_type(2)));
typedef float v8f __attribute__((ext_vector_type(8)));

// ---------------------------------------------------------------------------
// Fused  S = H·H^T  tile  ->  hinge-square loss  ->  per-tile partial sums.
// One workgroup = 64x64 output tile, 8 waves, each wave a 16x32 strip.
// ---------------------------------------------------------------------------
__global__ __launch_bounds__(256) void simloss_kernel(
    const float* __restrict__ bh, const int* __restrict__ labels,
    float* __restrict__ tilep)
{
    __shared__ float As[BT * STRIDE];   // 33792 B
    __shared__ float Bs[BT * STRIDE];   // 33792 B  (total ~66 KB of the 320 KB WGP LDS)

    const int tid = threadIdx.x;
    const int bi0 = blockIdx.y * BT;    // row block of S
    const int bj0 = blockIdx.x * BT;    // col block of S

    // Cooperative 128b loads: 64 rows x 128 f32 per matrix (8 float4 per thread).
    const float4* bh4 = (const float4*)bh;
    #pragma unroll
    for (int it = 0; it < 8; ++it) {
        int idx = it * 256 + tid;
        int r   = idx >> 5;             // 0..63
        int c4  = idx & 31;             // 0..31 (float4 column)
        float4 va = bh4[(size_t)(bi0 + r) * (KDIM / 4) + c4];
        float4 vb = bh4[(size_t)(bj0 + r) * (KDIM / 4) + c4];
        *(float4*)&As[r * STRIDE + c4 * 4] = va;
        *(float4*)&Bs[r * STRIDE + c4 * 4] = vb;
    }

    const int wave = tid >> 5;
    const int lane = tid & 31;
    const int wr   = wave >> 1;         // 0..3: which 16-row strip
    const int wc   = wave & 1;          // 0..1: which 32-col strip
    const int fr   = lane & 15;         // fragment row/col within 16
    const int koff = (lane >> 4) * 2;   // lanes 16..31 hold k+2,k+3

    // C/D layout: vgpr r -> M = r (+8 for lanes 16..31), N = lane&15.
    const int mbase = bi0 + wr * 16 + ((lane >> 4) << 3);
    const int nb0   = bj0 + wc * 32 + fr;

    // Warm the labels lines this thread consumes in the epilogue.
    #pragma unroll
    for (int r = 0; r < 8; ++r)
        __builtin_prefetch(&labels[(size_t)(mbase + r) * BDIM + nb0], 0, 0);

    __syncthreads();

    const float* arow  = &As[(wr * 16 + fr) * STRIDE + koff];
    const float* brow0 = &Bs[(wc * 32 + fr) * STRIDE + koff];
    const float* brow1 = brow0 + 16 * STRIDE;

    v8f acc0 = {};
    v8f acc1 = {};
    #pragma unroll 8
    for (int k0 = 0; k0 < KDIM; k0 += 4) {
        v2f a  = {arow[k0],  arow[k0 + 1]};
        v2f b0 = {brow0[k0], brow0[k0 + 1]};
        v2f b1 = {brow1[k0], brow1[k0 + 1]};
        acc0 = __builtin_amdgcn_wmma_f32_16x16x4_f32(
            false, a, false, b0, (short)0, acc0, false, false);
        acc1 = __builtin_amdgcn_wmma_f32_16x16x4_f32(
            false, a, false, b1, (short)0, acc1, false, false);
    }

    // Fused epilogue: masked squared-hinge terms + pos-pair count.
    float lsum = 0.f, lpos = 0.f;
    #pragma unroll
    for (int r = 0; r < 8; ++r) {
        const int i  = mbase + r;
        const int j0 = nb0;
        const int j1 = nb0 + 16;
        const int lab0 = labels[(size_t)i * BDIM + j0];
        const int lab1 = labels[(size_t)i * BDIM + j1];

        float s = acc0[r];
        float pt = fmaxf(0.9f - s, 0.f); pt *= pt;
        float nt = fmaxf(s + 0.1f, 0.f); nt *= nt;
        bool off0 = (i != j0);
        bool pos0 = off0 && (lab0 == 1);
        lsum += pos0 ? pt : (off0 ? nt : 0.f);
        lpos += pos0 ? 1.f : 0.f;

        s = acc1[r];
        pt = fmaxf(0.9f - s, 0.f); pt *= pt;
        nt = fmaxf(s + 0.1f, 0.f); nt *= nt;
        bool off1 = (i != j1);
        bool pos1 = off1 && (lab1 == 1);
        lsum += pos1 ? pt : (off1 ? nt : 0.f);
        lpos += pos1 ? 1.f : 0.f;
    }

    // Block reduction (reuse As as scratch), fixed order -> deterministic.
    __syncthreads();
    As[tid]       = lsum;
    As[256 + tid] = lpos;
    __syncthreads();
    for (int s = 128; s > 0; s >>= 1) {
        if (tid < s) {
            As[tid]       += As[tid + s];
            As[256 + tid] += As[256 + tid + s];
        }
        __syncthreads();
    }
    if (tid == 0) {
        const int t = blockIdx.y * gridDim.x + blockIdx.x;
        tilep[2 * t]     = As[0];
        tilep[2 * t + 1] = As[256];
    }
}

// ---------------------------------------------------------------------------
// quant partials: sum |  |x| - 1 |  over hash_logits (1M elems, grid-stride).
// ---------------------------------------------------------------------------
__global__ __launch_bounds__(256) void quant_kernel(
    const float* __restrict__ hl, float* __restrict__ qpart)
{
    __shared__ float red[256];
    const int tid = threadIdx.x;
    const int N = BDIM * KDIM;
    float acc = 0.f;
    for (int idx = blockIdx.x * 256 + tid; idx < N; idx += 256 * 256)
        acc += fabsf(fabsf(hl[idx]) - 1.0f);
    red[tid] = acc;
    __syncthreads();
    for (int s = 128; s > 0; s >>= 1) {
        if (tid < s) red[tid] += red[tid + s];
        __syncthreads();
    }
    if (tid == 0) qpart[blockIdx.x] = red[0];
}

// ---------------------------------------------------------------------------
// balance partials: per-block column sums of binary_hash (128 rows / block).
// ---------------------------------------------------------------------------
__global__ __launch_bounds__(256) void balance_kernel(
    const float* __restrict__ bh, float* __restrict__ balp)
{
    __shared__ float red[256];
    const int tid  = threadIdx.x;
    const int k    = tid & 127;
    const int half = tid >> 7;
    const int r0   = blockIdx.x * 128;
    float acc = 0.f;
    #pragma unroll 4
    for (int c = 0; c < 64; ++c) {
        int r = r0 + half + 2 * c;
        acc += bh[(size_t)r * KDIM + k];
    }
    red[tid] = acc;
    __syncthreads();
    if (tid < 128) balp[blockIdx.x * 128 + tid] = red[tid] + red[tid + 128];
}

// ---------------------------------------------------------------------------
// Finalize: fixed-order reduction of all partials -> scalar loss.
// ---------------------------------------------------------------------------
__global__ __launch_bounds__(256) void final_kernel(
    const float* __restrict__ tilep, const float* __restrict__ qpart,
    const float* __restrict__ balp, float* __restrict__ out)
{
    __shared__ float red[512];
    const int tid = threadIdx.x;

    float s_sim = 0.f, s_pos = 0.f;
    for (int t = tid; t < NTILE; t += 256) {
        s_sim += tilep[2 * t];
        s_pos += tilep[2 * t + 1];
    }
    red[tid]       = s_sim;
    red[256 + tid] = s_pos;
    __syncthreads();
    for (int s = 128; s > 0; s >>= 1) {
        if (tid < s) {
            red[tid]       += red[tid + s];
            red[256 + tid] += red[256 + tid + s];
        }
        __syncthreads();
    }
    const float sim_sum = red[0];
    const float pos_cnt = red[256];
    __syncthreads();

    red[tid] = qpart[tid];
    __syncthreads();
    for (int s = 128; s > 0; s >>= 1) {
        if (tid < s) red[tid] += red[tid + s];
        __syncthreads();
    }
    const float quant_sum = red[0];
    __syncthreads();

    float bal = 0.f;
    if (tid < 128) {
        float cs = 0.f;
        #pragma unroll 8
        for (int b = 0; b < 64; ++b) cs += balp[b * 128 + tid];
        float m = cs / (float)BDIM;
        bal = m * m;
    }
    red[tid] = bal;
    __syncthreads();
    for (int s = 128; s > 0; s >>= 1) {
        if (tid < s) red[tid] += red[tid + s];
        __syncthreads();
    }

    if (tid == 0) {
        float sim_loss = (pos_cnt > 0.f)
                             ? sim_sum / ((float)BDIM * (float)(BDIM - 1))
                             : sim_sum;
        float quant_loss = quant_sum / (float)(BDIM * KDIM);
        out[0] = 1.0f * sim_loss + 0.5f * quant_loss + 0.1f * red[0];
    }
}

// ---------------------------------------------------------------------------
extern "C" void kernel_launch(void* const* d_in, const int* in_sizes, int n_in,
                              void* d_out, int out_size, void* d_ws, size_t ws_size,
                              hipStream_t stream)
{
    (void)in_sizes; (void)n_in; (void)out_size; (void)ws_size;
    const float* hash_logits = (const float*)d_in[0];
    const float* binary_hash = (const float*)d_in[1];
    const int*   labels      = (const int*)d_in[2];

    float* ws    = (float*)d_ws;
    float* tilep = ws;                        // 2*NTILE  = 32768 floats
    float* qpart = ws + 2 * NTILE;            // 256 floats
    float* balp  = ws + 2 * NTILE + 256;      // 64*128   = 8192 floats

    dim3 grid(BDIM / BT, BDIM / BT);          // 128 x 128 tiles
    simloss_kernel<<<grid, 256, 0, stream>>>(binary_hash, labels, tilep);
    quant_kernel<<<256, 256, 0, stream>>>(hash_logits, qpart);
    balance_kernel<<<64, 256, 0, stream>>>(binary_hash, balp);
    final_kernel<<<1, 256, 0, stream>>>(tilep, qpart, balp, (float*)d_out);
}